// MIMS_19241453486098
// MI455X (gfx1250) — compile-verified
//
#include <hip/hip_runtime.h>
#include <math.h>

#define NP    110592          // 48^3 voxels
#define NSP   48
#define NSP2  2304            // 48*48
#define NB_SEQ 2304
#define L_SEQ 192
#define SEQ_PER_BLK 4
#define SEQ_STRIDE 3088       // 16 channels * 193 (padded) floats per sequence

typedef __attribute__((ext_vector_type(2))) float v2f;
typedef __attribute__((ext_vector_type(8))) float v8f;

typedef __attribute__((address_space(3))) int lds_int_t;
typedef __attribute__((address_space(1))) int glb_int_t;

__device__ __forceinline__ float silu_f(float v){ return v / (1.f + __expf(-v)); }

// CDNA5 async global->LDS copy (ASYNCcnt-tracked, no VGPR data); safe fallback.
__device__ __forceinline__ void async_cp_b32(const float* g, float* l){
#if defined(__has_builtin) && __has_builtin(__builtin_amdgcn_global_load_async_to_lds_b32)
  __builtin_amdgcn_global_load_async_to_lds_b32((glb_int_t*)g, (lds_int_t*)l, 0, 0);
#else
  *l = *g;
#endif
}
__device__ __forceinline__ void async_wait(){
#if defined(__has_builtin) && __has_builtin(__builtin_amdgcn_s_wait_asynccnt)
  __builtin_amdgcn_s_wait_asynccnt(0);
#endif
}

// ---------------- init ----------------
__global__ void init_stats_kernel(float* stats, int n){
  int i = blockIdx.x*blockDim.x + threadIdx.x;
  if (i < n) stats[i] = 0.f;
}

// ---------------- embed conv3d (k7 s2 p3) + stats ----------------
__global__ void embed_kernel(const float* __restrict__ x, const float* __restrict__ ew,
                             float* __restrict__ Uconv, float* __restrict__ estats){
  __shared__ float wsm[343];
  __shared__ float red[256];
  const int mc = blockIdx.y;              // m*16+c, 0..63
  const int m  = mc >> 4;
  for (int i = threadIdx.x; i < 343; i += 256) wsm[i] = ew[mc*343 + i];
  __syncthreads();
  const int vox = blockIdx.x*256 + threadIdx.x;        // 0..110591
  const int z = vox / NSP2; const int rem = vox - z*NSP2;
  const int y = rem / NSP;  const int xx = rem - y*NSP;
  const float* xin = x + (size_t)m * 96*96*96;
  float acc = 0.f;
  for (int kz = 0; kz < 7; ++kz){
    int iz = 2*z - 3 + kz;
    if ((unsigned)iz >= 96u) continue;
    for (int ky = 0; ky < 7; ++ky){
      int iy = 2*y - 3 + ky;
      if ((unsigned)iy >= 96u) continue;
      const float* row = xin + ((size_t)iz*96 + iy)*96;
      const float* wr  = wsm + kz*49 + ky*7;
      #pragma unroll
      for (int kx = 0; kx < 7; ++kx){
        int ix = 2*xx - 3 + kx;
        if ((unsigned)ix < 96u) acc += row[ix]*wr[kx];
      }
    }
  }
  Uconv[(size_t)mc*NP + vox] = acc;
  red[threadIdx.x] = acc; __syncthreads();
  for (int s = 128; s > 0; s >>= 1){
    if (threadIdx.x < s) red[threadIdx.x] += red[threadIdx.x + s];
    __syncthreads();
  }
  float bsum = red[0]; __syncthreads();
  red[threadIdx.x] = acc*acc; __syncthreads();
  for (int s = 128; s > 0; s >>= 1){
    if (threadIdx.x < s) red[threadIdx.x] += red[threadIdx.x + s];
    __syncthreads();
  }
  if (threadIdx.x == 0){
    atomicAdd(&estats[mc*2 + 0], bsum);
    atomicAdd(&estats[mc*2 + 1], red[0]);
  }
}

// ---------------- instance norm + exact GELU, init U_acc = U ----------------
__global__ void norm_gelu_kernel(const float* __restrict__ Uconv, const float* __restrict__ estats,
                                 float* __restrict__ U, float* __restrict__ Uacc){
  const int mc  = blockIdx.y;
  const int vox = blockIdx.x*256 + threadIdx.x;
  const float invn = 1.f / (float)NP;
  float mean = estats[mc*2 + 0] * invn;
  float var  = estats[mc*2 + 1] * invn - mean*mean;
  float v  = Uconv[(size_t)mc*NP + vox];
  float xn = (v - mean) * rsqrtf(var + 1e-5f);
  float g  = 0.5f * xn * (1.f + erff(xn * 0.70710678118654752f));
  U[(size_t)mc*NP + vox]    = g;
  Uacc[(size_t)mc*NP + vox] = g;
}

// ---------------- Mamba selective scan ----------------
// 4 waves/block, one wave32 per sequence; lane = d_inner index.
// Full sequence input staged to LDS (async-to-LDS when available) so the
// recurrent critical path never touches global memory.
__global__ void __launch_bounds__(128)
mamba_kernel(const float* __restrict__ U, float* __restrict__ Uacc,
             const float* __restrict__ inpw,  const float* __restrict__ convw,
             const float* __restrict__ convb, const float* __restrict__ xprojw,
             const float* __restrict__ dtw,   const float* __restrict__ dtb,
             const float* __restrict__ Alog,  const float* __restrict__ Dssm,
             const float* __restrict__ outw,  int axis){
  __shared__ float s_seq[SEQ_PER_BLK * SEQ_STRIDE]; // [seq][channel][t], stride 193
  __shared__ float s_inp[64*16];   // in_proj_w
  __shared__ float s_xp[33*32];    // x_proj_w: row0=dt, rows1..16=B, rows17..32=C
  __shared__ float s_out[16*32];   // out_proj_w
  const int tid = threadIdx.x;
  for (int i = tid; i < 1024; i += 128) s_inp[i] = inpw[i];
  for (int i = tid; i < 1056; i += 128) s_xp[i]  = xprojw[i];
  for (int i = tid; i < 512;  i += 128) s_out[i] = outw[i];

  // stage 4 sequences (4*64 rows of 48 elems) into LDS
  const int nb0 = blockIdx.x * SEQ_PER_BLK;
  for (int i = tid; i < SEQ_PER_BLK*64*NSP; i += 128){
    int s  = i / 3072; int r = i - s*3072;
    int mc = r / NSP;  int pos = r - mc*NSP;
    int m  = mc >> 4,  c = mc & 15;
    int nb = nb0 + s;
    int p0 = nb / NSP, p1 = nb - p0*NSP;
    int vox;
    if (axis == 0)      vox = p0*NSP2 + p1*NSP + pos;   // stride-1 in pos
    else if (axis == 1) vox = p0*NSP2 + pos*NSP + p1;
    else                vox = pos*NSP2 + p0*NSP + p1;
    // token index t = pos*4 + m ; channel-major padded LDS layout
    async_cp_b32(&U[(size_t)mc*NP + vox], &s_seq[s*SEQ_STRIDE + c*193 + pos*4 + m]);
  }
  async_wait();
  __syncthreads();

  const int lane = tid & 31;
  const int sq   = tid >> 5;                 // 0..3
  const int nb   = nb0 + sq;
  const int p0 = nb / NSP, p1 = nb - (nb/NSP)*NSP;
  const float* seqb = &s_seq[sq * SEQ_STRIDE];

  // per-lane weights straight from global (lane-indexed, tiny, L2-resident)
  float negA[16];
  #pragma unroll
  for (int n = 0; n < 16; ++n) negA[n] = -__expf(Alog[lane*16 + n]);
  const float cw0 = convw[lane*4+0], cw1 = convw[lane*4+1];
  const float cw2 = convw[lane*4+2], cw3 = convw[lane*4+3];
  const float cb  = convb[lane];
  const float dtwl = dtw[lane], dtbl = dtb[lane], Dl = Dssm[lane];

  float h[16];
  #pragma unroll
  for (int n = 0; n < 16; ++n) h[n] = 0.f;
  float win0 = 0.f, win1 = 0.f, win2 = 0.f;   // causal conv window (pad=3)

  for (int t = 0; t < L_SEQ; ++t){
    const int pos = t >> 2, m = t & 3;
    int z, y, xx;
    if (axis == 0)      { z = p0; y = p1;  xx = pos; }
    else if (axis == 1) { z = p0; y = pos; xx = p1;  }
    else                { z = pos; y = p0; xx = p1;  }
    const int voxbase = z*NSP2 + y*NSP + xx;

    // token input from LDS: lanes 0..15 each read one channel
    float seqv = (lane < 16) ? seqb[lane*193 + t] : 0.f;
    // in_proj rows d (xm) and d+32 (z gate)
    float xr = 0.f, zr = 0.f;
    #pragma unroll
    for (int c = 0; c < 16; ++c){
      float sc = __shfl(seqv, c, 32);
      xr += sc * s_inp[lane*16 + c];
      zr += sc * s_inp[(lane + 32)*16 + c];
    }
    // depthwise causal conv4 + silu
    float xc = cb + cw0*win0 + cw1*win1 + cw2*win2 + cw3*xr;
    win0 = win1; win1 = win2; win2 = xr;
    float xm = silu_f(xc);
    // dt_raw = <xm, x_proj_w[0]> via butterfly reduce
    float p = xm * s_xp[lane];
    #pragma unroll
    for (int off = 16; off >= 1; off >>= 1) p += __shfl_xor(p, off, 32);
    float dt_raw = p;
    // B (lanes 0..15 -> rows 1..16) / C (lanes 16..31 -> rows 17..32)
    float bc = 0.f;
    const int rowbc = 1 + lane;
    #pragma unroll
    for (int i = 0; i < 32; ++i) bc += __shfl(xm, i, 32) * s_xp[rowbc*32 + i];
    // dt = softplus(dt_raw * dtw + dtb)
    float dtv = dt_raw * dtwl + dtbl;
    dtv = (dtv > 20.f) ? dtv : log1pf(__expf(dtv));
    float dtx = dtv * xm;
    // selective-scan state update + output contraction over state
    float yacc = 0.f;
    #pragma unroll
    for (int n = 0; n < 16; ++n){
      float dA = __expf(dtv * negA[n]);
      float Bn = __shfl(bc, n, 32);
      float Cn = __shfl(bc, 16 + n, 32);
      h[n] = dA*h[n] + dtx*Bn;
      yacc += h[n]*Cn;
    }
    float yv = (yacc + xm*Dl) * silu_f(zr);
    // out_proj: lanes 0..15 each produce one output channel
    const int orow = lane & 15;
    float oc = 0.f;
    #pragma unroll
    for (int i = 0; i < 32; ++i) oc += __shfl(yv, i, 32) * s_out[orow*32 + i];
    if (lane < 16){
      size_t oi = (size_t)((m*16 + lane))*NP + voxbase;
      Uacc[oi] += oc;
    }
  }
}

// ---------------- fuse + res_fuse 1x1x1 convs as fp32 WMMA GEMMs ----------------
// Y[vox][oc] = sum_ch U[ch][vox] * W[oc][ch] ; K=64 in 16 chained 16x16x4 WMMAs.
__global__ void __launch_bounds__(256)
fuse_gemm_kernel(const float* __restrict__ Uacc, const float* __restrict__ Uorig,
                 const float* __restrict__ fw, const float* __restrict__ rw,
                 float* __restrict__ F, float* __restrict__ R){
  const int lane = threadIdx.x & 31;
  const int wv   = threadIdx.x >> 5;
  const int base = (blockIdx.x*8 + wv) * 128;     // 8 voxel tiles of 16 per wave
  const int nrow  = lane & 15;                    // N (out-channel) for B frag / D
  const int khalf = (lane >> 4) * 2;              // lanes 16-31 hold K=2,3
  v2f bF[16], bR[16];
  #pragma unroll
  for (int ks = 0; ks < 16; ++ks){
    int k0 = ks*4 + khalf;
    bF[ks].x = fw[nrow*64 + k0];  bF[ks].y = fw[nrow*64 + k0 + 1];
    bR[ks].x = rw[nrow*64 + k0];  bR[ks].y = rw[nrow*64 + k0 + 1];
  }
  const int mrow = lane & 15;                     // M (voxel-in-tile) for A frag
  for (int mt = 0; mt < 8; ++mt){
    const int vb = base + mt*16;
    v8f accF = {}; v8f accR = {};
    #pragma unroll
    for (int ks = 0; ks < 16; ++ks){
      int k0 = ks*4 + khalf;
      v2f aA; aA.x = Uacc[(size_t)k0*NP + vb + mrow];
              aA.y = Uacc[(size_t)(k0+1)*NP + vb + mrow];
      accF = __builtin_amdgcn_wmma_f32_16x16x4_f32(false, aA, false, bF[ks],
                                                   (short)0, accF, false, false);
      v2f aB; aB.x = Uorig[(size_t)k0*NP + vb + mrow];
              aB.y = Uorig[(size_t)(k0+1)*NP + vb + mrow];
      accR = __builtin_amdgcn_wmma_f32_16x16x4_f32(false, aB, false, bR[ks],
                                                   (short)0, accR, false, false);
    }
    const int voxrow = vb + (lane >> 4)*8;  // D: lanes 0-15 -> M=v, 16-31 -> M=v+8
    #pragma unroll
    for (int v = 0; v < 8; ++v){
      F[(size_t)nrow*NP + voxrow + v] = accF[v];
      R[(size_t)nrow*NP + voxrow + v] = accR[v];
    }
  }
}

// ---------------- per-out-channel stats of F ----------------
__global__ void fstats_kernel(const float* __restrict__ F, float* __restrict__ fstats){
  __shared__ float s1[256], s2[256];
  const int oc = blockIdx.y;
  float sum = 0.f, sq = 0.f;
  for (int i = blockIdx.x*256 + threadIdx.x; i < NP; i += gridDim.x*256){
    float v = F[(size_t)oc*NP + i]; sum += v; sq += v*v;
  }
  s1[threadIdx.x] = sum; s2[threadIdx.x] = sq; __syncthreads();
  for (int s = 128; s > 0; s >>= 1){
    if (threadIdx.x < s){ s1[threadIdx.x] += s1[threadIdx.x+s]; s2[threadIdx.x] += s2[threadIdx.x+s]; }
    __syncthreads();
  }
  if (threadIdx.x == 0){
    atomicAdd(&fstats[oc*2 + 0], s1[0]);
    atomicAdd(&fstats[oc*2 + 1], s2[0]);
  }
}

// ---------------- final: inorm(F) + R ----------------
__global__ void final_kernel(const float* __restrict__ F, const float* __restrict__ R,
                             const float* __restrict__ fstats, float* __restrict__ out){
  const int oc = blockIdx.y;
  const int i  = blockIdx.x*256 + threadIdx.x;
  const float invn = 1.f / (float)NP;
  float mean = fstats[oc*2 + 0] * invn;
  float var  = fstats[oc*2 + 1] * invn - mean*mean;
  out[(size_t)oc*NP + i] = (F[(size_t)oc*NP + i] - mean) * rsqrtf(var + 1e-5f)
                          + R[(size_t)oc*NP + i];
}

extern "C" void kernel_launch(void* const* d_in, const int* in_sizes, int n_in,
                              void* d_out, int out_size, void* d_ws, size_t ws_size,
                              hipStream_t stream){
  const float* x        = (const float*)d_in[0];
  const float* embed_w  = (const float*)d_in[1];
  const float* in_proj  = (const float*)d_in[2];
  const float* conv1d_w = (const float*)d_in[3];
  const float* conv1d_b = (const float*)d_in[4];
  const float* x_proj_w = (const float*)d_in[5];
  const float* dt_proj_w= (const float*)d_in[6];
  const float* dt_proj_b= (const float*)d_in[7];
  const float* A_log    = (const float*)d_in[8];
  const float* D_ssm    = (const float*)d_in[9];
  const float* out_proj = (const float*)d_in[10];
  const float* fuse_w   = (const float*)d_in[11];
  const float* res_fuse = (const float*)d_in[12];
  float* out = (float*)d_out;

  const size_t SEG = (size_t)64 * NP * sizeof(float);   // 28,311,552 B
  char* ws = (char*)d_ws;
  float* Uconv = (float*)(ws);
  float* U     = (float*)(ws + SEG);
  float* Uacc  = (float*)(ws + 2*SEG);
  float* stats = (float*)(ws + 3*SEG);   // 128 embed stats + 32 fuse stats
  float* estats = stats;
  float* fstats = stats + 128;
  float* F = Uconv;                       // Uconv dead after norm_gelu: reuse
  float* Rb = Uconv + (size_t)16*NP;

  init_stats_kernel<<<1, 256, 0, stream>>>(stats, 160);
  embed_kernel<<<dim3(NP/256, 64), 256, 0, stream>>>(x, embed_w, Uconv, estats);
  norm_gelu_kernel<<<dim3(NP/256, 64), 256, 0, stream>>>(Uconv, estats, U, Uacc);
  for (int axis = 0; axis < 3; ++axis){
    mamba_kernel<<<NB_SEQ/SEQ_PER_BLK, 128, 0, stream>>>(U, Uacc, in_proj, conv1d_w, conv1d_b,
                                                         x_proj_w, dt_proj_w, dt_proj_b,
                                                         A_log, D_ssm, out_proj, axis);
  }
  fuse_gemm_kernel<<<NP/1024, 256, 0, stream>>>(Uacc, U, fuse_w, res_fuse, F, Rb);
  fstats_kernel<<<dim3(32, 16), 256, 0, stream>>>(F, fstats);
  final_kernel<<<dim3(NP/256, 16), 256, 0, stream>>>(F, Rb, fstats, out);
}